// Model_45363444580560
// MI455X (gfx1250) — compile-verified
//
#include <hip/hip_runtime.h>
#include <math.h>

typedef __attribute__((ext_vector_type(2))) float v2f;
typedef __attribute__((ext_vector_type(8))) float v8f;

#define WPB 4  // waves per block (4 x wave32 = 128 threads)

__device__ __forceinline__ float gelu_exact(float x) {
    return 0.5f * x * (1.0f + erff(x * 0.70710678118654752f));
}

// ---------------------------------------------------------------------------
// Generic fp32 WMMA GEMM, one wave per 16x16 output tile.
//   BT=false: C = A[M,K] (row-major) x B[K,N] (row-major)
//   BT=true : C = A[M,K] (row-major) x B[N,K]^T (B row-major [N,K], "X @ W^T")
// Batched via strides; optional split-K (partials written per k-chunk);
// epilogue (only when kChunks==1): + bias[n], + addScale*addSrc, optional GELU.
// Out-of-range M/N lanes load CLAMPED (in-bounds) addresses; the garbage they
// accumulate lives only in accumulator elements that are never stored, so the
// hot loop needs no EXEC-mask juggling and software-pipelines cleanly.
// Fragment layouts per CDNA5 ISA 7.12.2:
//   A (16x4 f32, v2f): lane l -> row M=l&15, VGPR j -> K = k + 2*(l>>4) + j
//   B (4x16 f32, v2f): lane l -> col N=l&15, VGPR j -> K = k + 2*(l>>4) + j
//   C/D (16x16 f32, v8f): VGPR r, lane l -> M = r + 8*(l>>4), N = l&15
// ---------------------------------------------------------------------------
template <bool BT>
__global__ void gemm_wmma_f32(const float* __restrict__ A,
                              const float* __restrict__ B,
                              float* __restrict__ C,
                              int M, int N, int K,
                              long sA, long sB, long sC,
                              int lda, int ldb, int ldc,
                              int kChunks, long chunkStride,
                              const float* __restrict__ bias,
                              const float* __restrict__ addSrc,
                              float addScale, long sAdd,
                              int actGelu)
{
    const int lane = threadIdx.x & 31;
    const int wave = threadIdx.x >> 5;
    const int MT = (M + 15) >> 4;
    const int NT = (N + 15) >> 4;
    const int tile = blockIdx.x * WPB + wave;
    if (tile >= MT * NT) return;               // wave-uniform exit
    const int mt = tile / NT;
    const int nt = tile - mt * NT;
    const int b  = blockIdx.z;

    const float* Ab = A + (long)b * sA;
    const float* Bb = B + (long)b * sB;

    const int half = lane >> 4;                // 0|1
    const int l16  = lane & 15;
    const int h2   = half * 2;                 // k sub-offset 0|2
    const int mrow = mt * 16 + l16;            // A row handled by this lane
    const int ncol = nt * 16 + l16;            // B column / store column
    const int mldr = (mrow < M) ? mrow : (M - 1);  // clamped load row
    const int nldr = (ncol < N) ? ncol : (N - 1);  // clamped load col

    const int kPer = K / kChunks;
    const int k0 = kPer * (int)blockIdx.y;
    const int k1 = k0 + kPer;

    v8f acc = {};
    const float* Aptr = Ab + (long)mldr * lda + h2;
    const float* Bptr = BT ? (Bb + (long)nldr * ldb + h2)   // row of B[N,K]
                           : (Bb + nldr);                    // col of B[K,N]

    int k = k0;
    // main loop: 16 K per trip, 8 loads in flight, then 4 WMMAs
    for (; k + 16 <= k1; k += 16) {
        float2 ta[4], tb[4];
        if (BT) __builtin_prefetch((const void*)(Bptr + k + 256), 0, 0);
#pragma unroll
        for (int u = 0; u < 4; ++u) {
            ta[u] = *(const float2*)(Aptr + k + 4 * u);
            if (BT) {
                tb[u] = *(const float2*)(Bptr + k + 4 * u);
            } else {
                const float* bp = Bptr + (long)(k + 4 * u + h2) * ldb;
                tb[u].x = bp[0];
                tb[u].y = bp[ldb];
            }
        }
#pragma unroll
        for (int u = 0; u < 4; ++u) {
            v2f a, bf;
            a.x = ta[u].x; a.y = ta[u].y;
            bf.x = tb[u].x; bf.y = tb[u].y;
            acc = __builtin_amdgcn_wmma_f32_16x16x4_f32(
                false, a, false, bf, (short)0, acc, false, false);
        }
    }
    // tail: 4 K per trip (covers K % 16, e.g. fc2 K=1000, fc3 K=100)
    for (; k < k1; k += 4) {
        float2 ta = *(const float2*)(Aptr + k);
        float2 tb;
        if (BT) {
            tb = *(const float2*)(Bptr + k);
        } else {
            const float* bp = Bptr + (long)(k + h2) * ldb;
            tb.x = bp[0];
            tb.y = bp[ldb];
        }
        v2f a, bf;
        a.x = ta.x; a.y = ta.y;
        bf.x = tb.x; bf.y = tb.y;
        acc = __builtin_amdgcn_wmma_f32_16x16x4_f32(
            false, a, false, bf, (short)0, acc, false, false);
    }

    float* Cb = C + (long)b * sC + (long)blockIdx.y * chunkStride;
    const float* addB = addSrc ? (addSrc + (long)b * sAdd) : nullptr;
    if (ncol < N) {
#pragma unroll
        for (int r = 0; r < 8; ++r) {
            int mr = mt * 16 + r + half * 8;
            if (mr < M) {
                float v = acc[r];
                if (kChunks == 1) {
                    if (bias)    v += bias[ncol];
                    if (addB)    v += addScale * addB[(long)mr * ldc + ncol];
                    if (actGelu) v = gelu_exact(v);
                }
                Cb[(long)mr * ldc + ncol] = v;
            }
        }
    }
}

// ---------------------------------------------------------------------------
// Support kernels
// ---------------------------------------------------------------------------
__global__ void row_sumsq_kernel(const float* __restrict__ X,
                                 float* __restrict__ out, int ncols)
{
    __shared__ float red[256];
    const int row = blockIdx.x;
    const float* xr = X + (long)row * ncols;
    float s = 0.0f;
    for (int c = threadIdx.x; c < ncols; c += blockDim.x) {
        float v = xr[c];
        s += v * v;
    }
    red[threadIdx.x] = s;
    __syncthreads();
    for (int off = blockDim.x >> 1; off > 0; off >>= 1) {
        if ((int)threadIdx.x < off) red[threadIdx.x] += red[threadIdx.x + off];
        __syncthreads();
    }
    if (threadIdx.x == 0) out[row] = red[0];
}

__global__ void dw_kernel(const float* __restrict__ alpha, float* __restrict__ dw)
{
    const int i = blockIdx.x, j = threadIdx.x;   // 256 x 256
    dw[i * 256 + j] = expf(-alpha[0] * fabsf((float)(i - j)));
}

// buf holds dot(E,L) on entry; produces prob (row-normalized gaussian sim).
__global__ void prob_kernel(float* __restrict__ buf, float* __restrict__ prob,
                            const float* __restrict__ enorm,
                            const float* __restrict__ lnorm)
{
    const int r = blockIdx.x * blockDim.x + threadIdx.x;  // 0..2047 = b*256+n
    if (r >= 8 * 256) return;
    const float e = enorm[r];
    float* row = buf + (long)r * 64;
    float mn = 3.4e38f, mx = -3.4e38f;
    for (int k = 0; k < 64; ++k) {
        float d2 = e + lnorm[k] - 2.0f * row[k];
        float d  = sqrtf(fmaxf(d2, 0.0f));
        row[k] = d;
        mn = fminf(mn, d);
        mx = fmaxf(mx, d);
    }
    const float inv = 1.0f / (mx - mn + 1e-6f);
    const float cc  = 1.0f / (2.0f * 0.3f * 0.3f);
    float sum = 0.0f;
    for (int k = 0; k < 64; ++k) {
        float nd = (row[k] - mn) * inv;
        float s  = expf(-nd * nd * cc);
        row[k] = s;
        sum += s;
    }
    const float isum = 1.0f / sum;
    float* pr = prob + (long)r * 64;
    for (int k = 0; k < 64; ++k) pr[k] = row[k] * isum;
}

__global__ void transpose_prob_kernel(const float* __restrict__ prob,
                                      float* __restrict__ Pt)
{
    // prob [8][256][64] -> Pt [8][64][256]
    const int idx = blockIdx.x * blockDim.x + threadIdx.x;
    if (idx >= 8 * 64 * 256) return;
    const int b   = idx / (64 * 256);
    const int rem = idx - b * 64 * 256;
    const int k   = rem / 256;
    const int n   = rem - k * 256;
    Pt[idx] = prob[(long)b * 256 * 64 + (long)n * 64 + k];
}

__global__ void adj_norm_kernel(const float* __restrict__ G, float* __restrict__ An)
{
    __shared__ float dinv[64];
    const int b = blockIdx.x, i = threadIdx.x;   // 8 blocks x 64 threads
    const float* Gb = G + (long)b * 4096;
    float deg = 0.0f;
    for (int j = 0; j < 64; ++j) {
        float a = (Gb[i * 64 + j] != 0.0f || i == j) ? 1.0f : 0.0f;
        deg += a;
    }
    dinv[i] = (deg > 0.0f) ? (1.0f / sqrtf(deg)) : 0.0f;
    __syncthreads();
    const float di = dinv[i];
    float* Ab = An + (long)b * 4096;
    for (int j = 0; j < 64; ++j) {
        float a = (Gb[i * 64 + j] != 0.0f || i == j) ? 1.0f : 0.0f;
        Ab[i * 64 + j] = a * di * dinv[j];
    }
}

__global__ void fc1_reduce_gelu_kernel(const float* __restrict__ part,
                                       const float* __restrict__ bias,
                                       float* __restrict__ out, int chunks)
{
    const int MN = 8 * 1000;
    const int idx = blockIdx.x * blockDim.x + threadIdx.x;
    if (idx >= MN) return;
    float s = 0.0f;
    for (int c = 0; c < chunks; ++c) s += part[(long)c * MN + idx];
    s += bias[idx % 1000];
    out[idx] = gelu_exact(s);
}

// ---------------------------------------------------------------------------
// Host-side launch helper
// ---------------------------------------------------------------------------
static void launch_gemm(bool bt, const float* A, const float* B, float* C,
                        int M, int N, int K,
                        long sA, long sB, long sC,
                        int lda, int ldb, int ldc,
                        int batches, int kChunks, long chunkStride,
                        const float* bias, const float* addSrc,
                        float addScale, long sAdd, int actGelu,
                        hipStream_t stream)
{
    const int MT = (M + 15) / 16, NT = (N + 15) / 16;
    const int tiles = MT * NT;
    dim3 grid((tiles + WPB - 1) / WPB, kChunks, batches);
    dim3 block(32 * WPB, 1, 1);
    if (bt)
        gemm_wmma_f32<true><<<grid, block, 0, stream>>>(
            A, B, C, M, N, K, sA, sB, sC, lda, ldb, ldc,
            kChunks, chunkStride, bias, addSrc, addScale, sAdd, actGelu);
    else
        gemm_wmma_f32<false><<<grid, block, 0, stream>>>(
            A, B, C, M, N, K, sA, sB, sC, lda, ldb, ldc,
            kChunks, chunkStride, bias, addSrc, addScale, sAdd, actGelu);
}

extern "C" void kernel_launch(void* const* d_in, const int* in_sizes, int n_in,
                              void* d_out, int out_size, void* d_ws, size_t ws_size,
                              hipStream_t stream)
{
    const float* E     = (const float*)d_in[0];   // [8,256,768]
    const float* L     = (const float*)d_in[1];   // [64,768]
    const float* alpha = (const float*)d_in[2];   // [1]
    const float* w1    = (const float*)d_in[3];   // [768,768]
    const float* b1    = (const float*)d_in[4];
    const float* w2    = (const float*)d_in[5];
    const float* b2    = (const float*)d_in[6];
    const float* w3    = (const float*)d_in[7];
    const float* b3    = (const float*)d_in[8];
    const float* fc1w  = (const float*)d_in[9];   // [1000,49152]
    const float* fc1b  = (const float*)d_in[10];
    const float* fc2w  = (const float*)d_in[11];  // [100,1000]
    const float* fc2b  = (const float*)d_in[12];
    const float* fc3w  = (const float*)d_in[13];  // [10,100]
    const float* fc3b  = (const float*)d_in[14];
    float* out = (float*)d_out;                    // [8,10]

    // Workspace carving (floats)
    float* ws    = (float*)d_ws;
    float* lnorm = ws; ws += 64;
    float* enorm = ws; ws += 2048;
    float* dw    = ws; ws += 256 * 256;
    float* dotb  = ws; ws += 2048 * 64;            // dot -> dist -> sim scratch
    float* prob  = ws; ws += 2048 * 64;            // [8][256][64]
    float* Pt    = ws; ws += 8 * 64 * 256;         // [8][64][256]
    float* T1    = ws; ws += 8 * 64 * 256;
    float* G     = ws; ws += 8 * 64 * 64;
    float* An    = ws; ws += 8 * 64 * 64;
    float* pool  = ws; ws += 8 * 64 * 768;         // cluster pooling "out"
    float* Y     = ws; ws += 8 * 64 * 768;         // hx @ W^T
    float* hx    = ws; ws += 8 * 64 * 768;         // GCN state / final x
    float* part  = ws; ws += 64 * 8 * 1000;        // fc1 split-K partials
    float* x1    = ws; ws += 8 * 1000;
    float* x2    = ws; ws += 8 * 100;

    // 1) squared row norms
    row_sumsq_kernel<<<64, 256, 0, stream>>>(L, lnorm, 768);
    row_sumsq_kernel<<<2048, 256, 0, stream>>>(E, enorm, 768);
    // 2) distance-weight matrix dw[256,256]
    dw_kernel<<<256, 256, 0, stream>>>(alpha, dw);
    // 3) dot = E[2048,768] @ L^T  (NT)
    launch_gemm(true, E, L, dotb, 2048, 64, 768, 0, 0, 0, 768, 768, 64,
                1, 1, 0, nullptr, nullptr, 0.0f, 0, 0, stream);
    // 4) prob (dist -> min/max norm -> gaussian -> row-normalize)
    prob_kernel<<<8, 256, 0, stream>>>(dotb, prob, enorm, lnorm);
    // 5) Pt = prob^T per batch
    transpose_prob_kernel<<<(8 * 64 * 256 + 255) / 256, 256, 0, stream>>>(prob, Pt);
    // 6) T1 = Pt @ dw  (NN, dw shared across batches)
    launch_gemm(false, Pt, dw, T1, 64, 256, 256, 64 * 256, 0, 64 * 256,
                256, 256, 256, 8, 1, 0, nullptr, nullptr, 0.0f, 0, 0, stream);
    // 7) G = T1 @ prob_b  (NN)  -> [8][64][64]
    launch_gemm(false, T1, prob, G, 64, 64, 256, 64 * 256, 256 * 64, 64 * 64,
                256, 64, 64, 8, 1, 0, nullptr, nullptr, 0.0f, 0, 0, stream);
    // 8) An = D^-1/2 (G!=0 | I) D^-1/2
    adj_norm_kernel<<<8, 64, 0, stream>>>(G, An);
    // 9) pool = Pt @ E_b  (NN) -> [8][64][768]
    launch_gemm(false, Pt, E, pool, 64, 768, 256, 64 * 256, 256 * 768, 64 * 768,
                256, 768, 768, 8, 1, 0, nullptr, nullptr, 0.0f, 0, 0, stream);
    // 10) GCN layer 1: Y = pool @ w1^T ; hx = An @ Y + b1
    launch_gemm(true, pool, w1, Y, 64, 768, 768, 64 * 768, 0, 64 * 768,
                768, 768, 768, 8, 1, 0, nullptr, nullptr, 0.0f, 0, 0, stream);
    launch_gemm(false, An, Y, hx, 64, 768, 64, 64 * 64, 64 * 768, 64 * 768,
                64, 768, 768, 8, 1, 0, b1, nullptr, 0.0f, 0, 0, stream);
    // 11) GCN layer 2
    launch_gemm(true, hx, w2, Y, 64, 768, 768, 64 * 768, 0, 64 * 768,
                768, 768, 768, 8, 1, 0, nullptr, nullptr, 0.0f, 0, 0, stream);
    launch_gemm(false, An, Y, hx, 64, 768, 64, 64 * 64, 64 * 768, 64 * 768,
                64, 768, 768, 8, 1, 0, b2, nullptr, 0.0f, 0, 0, stream);
    // 12) GCN layer 3 with fused residual: hx = An@Y + b3 + 2*pool  (== x)
    launch_gemm(true, hx, w3, Y, 64, 768, 768, 64 * 768, 0, 64 * 768,
                768, 768, 768, 8, 1, 0, nullptr, nullptr, 0.0f, 0, 0, stream);
    launch_gemm(false, An, Y, hx, 64, 768, 64, 64 * 64, 64 * 768, 64 * 768,
                64, 768, 768, 8, 1, 0, b3, pool, 2.0f, 64 * 768, 0, stream);
    // 13) fc1: split-K=64 over K=49152 (streams 196MB of fc1_w across 4032 waves)
    launch_gemm(true, hx, fc1w, part, 8, 1000, 49152, 0, 0, 0,
                49152, 49152, 1000, 1, 64, (long)8 * 1000,
                nullptr, nullptr, 0.0f, 0, 0, stream);
    fc1_reduce_gelu_kernel<<<(8 * 1000 + 255) / 256, 256, 0, stream>>>(part, fc1b, x1, 64);
    // 14) fc2 (+bias +GELU fused)
    launch_gemm(true, x1, fc2w, x2, 8, 100, 1000, 0, 0, 0, 1000, 1000, 100,
                1, 1, 0, fc2b, nullptr, 0.0f, 0, 1, stream);
    // 15) fc3 (+bias) -> d_out [8,10]
    launch_gemm(true, x2, fc3w, out, 8, 10, 100, 0, 0, 0, 100, 100, 10,
                1, 1, 0, fc3b, nullptr, 0.0f, 0, 0, stream);
}